// Pairwise_Interaction_Layer_75531294867457
// MI455X (gfx1250) — compile-verified
//
#include <hip/hip_runtime.h>

// Pairwise interaction layer: out[b, p, e] = in[b, I[p], e] * in[b, J[p], e]
// for the strict upper triangle (i<j) of a 39-field, 64-dim embedding table.
// Pure bandwidth problem (~818 MB traffic, ~0.19 GFLOP): stage each input row
// in LDS via CDNA5 async global->LDS DMA (ASYNCcnt), build the pair index
// table once per block, then stream 128-bit non-temporal stores.

#define NUM_FIELD 39
#define EMBED     64
#define NPAIR     741                      // 39*38/2, row-major triu order
#define BLOCK     256                      // 8 wave32s
#define ROW_F4    ((NUM_FIELD * EMBED) / 4)  // 624 float4 per input row
#define PSTRIDE   (BLOCK / 16)             // 16 pairs in flight per block pass

typedef float v4f __attribute__((ext_vector_type(4)));
// Exact pointee type of the async-copy builtins (from compiler diagnostic):
// 'int __attribute__((vector_size(16)))' in addrspace(1) / addrspace(3).
typedef int v4i_vs __attribute__((vector_size(16)));
typedef __attribute__((address_space(1))) v4i_vs g_v4i;   // global
typedef __attribute__((address_space(3))) v4i_vs l_v4i;   // LDS

#if defined(__gfx1250__)
#define HAS_ASYNC_LDS __has_builtin(__builtin_amdgcn_global_load_async_to_lds_b128)
#else
#define HAS_ASYNC_LDS 0
#endif

__global__ __launch_bounds__(BLOCK)
void pairwise_interaction_kernel(const float* __restrict__ in,
                                 float* __restrict__ out) {
    __shared__ float    tile[NUM_FIELD * EMBED];  // 9984 B: one batch row
    __shared__ unsigned pairOff[NPAIR];           // packed (i*E) | (j*E)<<16

    const int    tid = threadIdx.x;
    const size_t b   = blockIdx.x;
    const float* src = in + b * (size_t)(NUM_FIELD * EMBED);

    // ---- Stage input row b into LDS (async DMA, 16B/lane) ---------------
#if HAS_ASYNC_LDS
    for (int v = tid; v < ROW_F4; v += BLOCK) {
        __builtin_amdgcn_global_load_async_to_lds_b128(
            (g_v4i*)(src + v * 4),
            (l_v4i*)(tile + v * 4),
            /*offset=*/0, /*cpol=*/0);
    }
#else
    for (int v = tid; v < ROW_F4; v += BLOCK) {
        ((v4f*)tile)[v] = ((const v4f*)src)[v];   // global_load_b128 + ds_store_b128
    }
#endif

    // ---- Build pair table (overlaps the async copies in flight) ---------
    // Row-major triu(k=1): pairs with first index i start at
    // offset(i) = i*(F-1) - i*(i-1)/2 and have j = i+1 .. F-1.
    if (tid < NUM_FIELD - 1) {
        const int      i     = tid;
        int            off   = i * (NUM_FIELD - 1) - (i * (i - 1)) / 2;
        const unsigned ibase = (unsigned)(i * EMBED);
        for (int j = i + 1; j < NUM_FIELD; ++j)
            pairOff[off++] = ibase | ((unsigned)(j * EMBED) << 16);
    }

#if HAS_ASYNC_LDS
#if __has_builtin(__builtin_amdgcn_s_wait_asynccnt)
    __builtin_amdgcn_s_wait_asynccnt(0);
#else
    asm volatile("s_wait_asynccnt 0" ::: "memory");
#endif
#endif
    __syncthreads();

    // ---- Stream outputs -------------------------------------------------
    // lane layout: c = 16B chunk within the 256B output row (16 chunks),
    // 16 pairs in flight per block pass; a wave writes 2 consecutive pairs
    // = 512 contiguous bytes per pass. Unroll x4 so each thread keeps
    // 4 table loads + 8 ds_load_b128 + 4 NT global_store_b128 in flight.
    const int    c  = (tid & 15) * 4;              // float offset within row
    const int    p0 = tid >> 4;                    // first pair for this thread
    float*       op = out + b * (size_t)(NPAIR * EMBED) + p0 * EMBED + c;

    int p = p0;
#pragma unroll 1
    for (; p + 3 * PSTRIDE < NPAIR; p += 4 * PSTRIDE, op += 4 * PSTRIDE * EMBED) {
#pragma unroll
        for (int u = 0; u < 4; ++u) {
            const unsigned po = pairOff[p + u * PSTRIDE];
            const v4f a  = *(const v4f*)&tile[(po & 0xFFFFu) + c];  // ds_load_b128
            const v4f bb = *(const v4f*)&tile[(po >> 16)     + c];  // ds_load_b128
            __builtin_nontemporal_store(a * bb, (v4f*)(op + u * PSTRIDE * EMBED));
        }
    }
    // Tail: NPAIR = 741 -> 46 full passes of 16 pairs + 5 leftover pairs
#pragma unroll 1
    for (; p < NPAIR; p += PSTRIDE, op += PSTRIDE * EMBED) {
        const unsigned po = pairOff[p];
        const v4f a  = *(const v4f*)&tile[(po & 0xFFFFu) + c];
        const v4f bb = *(const v4f*)&tile[(po >> 16)     + c];
        __builtin_nontemporal_store(a * bb, (v4f*)op);
    }
}

extern "C" void kernel_launch(void* const* d_in, const int* in_sizes, int n_in,
                              void* d_out, int out_size, void* d_ws, size_t ws_size,
                              hipStream_t stream) {
    const float* in  = (const float*)d_in[0];
    float*       out = (float*)d_out;
    const int    batch = in_sizes[0] / (NUM_FIELD * EMBED);   // 4096
    pairwise_interaction_kernel<<<batch, BLOCK, 0, stream>>>(in, out);
}